// BatchSuperQMulti_62311385530811
// MI455X (gfx1250) — compile-verified
//
#include <hip/hip_runtime.h>
#include <cmath>

typedef float v2f __attribute__((ext_vector_type(2)));
typedef float v8f __attribute__((ext_vector_type(8)));

#define SQ_EPS 1e-6f
#define SQ_BIG 1e10f

__device__ __forceinline__ float sgnf(float v)   { return v > 0.0f ? 1.0f : -1.0f; }
__device__ __forceinline__ float clipEB(float v) { return fminf(fmaxf(v, SQ_EPS), SQ_BIG); }
// pow for strictly-positive NORMAL base (guaranteed in [1e-6, 1e10] by clipEB):
// raw hardware v_log_f32 / v_exp_f32, no libm denormal guards needed.
__device__ __forceinline__ float powp(float b, float p) {
    return __builtin_amdgcn_exp2f(p * __builtin_amdgcn_logf(b));
}
// fast reciprocal: single v_rcp_f32 (divisor is guaranteed |v| >= EPS, finite)
__device__ __forceinline__ float frcp(float v)   { return __builtin_amdgcn_rcpf(v); }

#define NPRIM_MAX 32
#define PSTRIDE   24
// prm[n] layout:
//  [0..11] affine rows of R^T with translation folded:  row r -> {r*4+0..r*4+3}
//  [12] 1/sx  [13] 1/sy  [14] 1/sz  [15] kx/sz  [16] ky/sz
//  [17] 1/e2  [18] 1/e1  [19] e2/e1 [20] -e1/2  [21] exist(1/0)

__global__ void __launch_bounds__(256)
sq_sdf_kernel(const float* __restrict__ points,
              const float* __restrict__ raw_scale,
              const float* __restrict__ raw_exponents,
              const float* __restrict__ raw_rotation,
              const float* __restrict__ translation,
              const float* __restrict__ raw_tapering,
              const int*   __restrict__ exist_mask,
              float* __restrict__ out,
              int M, int N, int tilesPerBatch)
{
    __shared__ float prm[NPRIM_MAX][PSTRIDE];
    const int b = blockIdx.y;
    const int t = threadIdx.x;

    // ---- per-primitive preprocessing (threads 0..N-1) ----
    if (t < N) {
        const int gi = b * N + t;
        float sx = expf(raw_scale[gi*3+0]) + 0.01f;
        float sy = expf(raw_scale[gi*3+1]) + 0.01f;
        float sz = expf(raw_scale[gi*3+2]) + 0.01f;
        float e1 = 1.0f/(1.0f + expf(-raw_exponents[gi*2+0])) * 1.8f + 0.1f;
        float e2 = 1.0f/(1.0f + expf(-raw_exponents[gi*2+1])) * 1.8f + 0.1f;
        float kx = tanhf(raw_tapering[gi*2+0]);
        float ky = tanhf(raw_tapering[gi*2+1]);
        float qw = raw_rotation[gi*4+0], qx = raw_rotation[gi*4+1];
        float qy = raw_rotation[gi*4+2], qz = raw_rotation[gi*4+3];
        float qn = rsqrtf(qw*qw + qx*qx + qy*qy + qz*qz);
        qw *= qn; qx *= qn; qy *= qn; qz *= qn;
        float R00 = 1.f-2.f*(qy*qy+qz*qz), R01 = 2.f*(qx*qy-qw*qz), R02 = 2.f*(qx*qz+qw*qy);
        float R10 = 2.f*(qx*qy+qw*qz), R11 = 1.f-2.f*(qx*qx+qz*qz), R12 = 2.f*(qy*qz-qw*qx);
        float R20 = 2.f*(qx*qz-qw*qy), R21 = 2.f*(qy*qz+qw*qx), R22 = 1.f-2.f*(qx*qx+qy*qy);
        float tx = translation[gi*3+0], ty = translation[gi*3+1], tz = translation[gi*3+2];
        // X_i = sum_j R[j][i]*p_j - sum_j R[j][i]*t_j   (einsum 'bnji,bnjm')
        prm[t][0] = R00; prm[t][1] = R10; prm[t][2]  = R20; prm[t][3]  = -(R00*tx + R10*ty + R20*tz);
        prm[t][4] = R01; prm[t][5] = R11; prm[t][6]  = R21; prm[t][7]  = -(R01*tx + R11*ty + R21*tz);
        prm[t][8] = R02; prm[t][9] = R12; prm[t][10] = R22; prm[t][11] = -(R02*tx + R12*ty + R22*tz);
        prm[t][12] = 1.f/sx; prm[t][13] = 1.f/sy; prm[t][14] = 1.f/sz;
        prm[t][15] = kx/sz;  prm[t][16] = ky/sz;
        prm[t][17] = 1.f/e2; prm[t][18] = 1.f/e1;
        prm[t][19] = e2/e1;  prm[t][20] = -0.5f*e1;
        prm[t][21] = (exist_mask[gi] != 0) ? 1.0f : 0.0f;
    }
    __syncthreads();

    const int lane = threadIdx.x & 31;
    const int wave = threadIdx.x >> 5;
    const int wavesPerBlock = blockDim.x >> 5;
    const int waveStride = gridDim.x * wavesPerBlock;
    const int m15 = lane & 15;          // point slot / D column
    const int hi  = lane >> 4;          // 0: K=0,1 half ; 1: K=2,3 half
    const int r   = m15 & 7;            // affine row within half
    const float rowMask = (r < 3) ? 1.0f : 0.0f;
    const int   rc      = (r < 3) ? r : 0;
    const int   aoff    = rc * 4 + 2 * hi;   // which 2 of the 4 affine cols this lane carries
    const int nPairs = N >> 1;

    for (int tile = blockIdx.x * wavesPerBlock + wave; tile < tilesPerBatch; tile += waveStride) {
        int pt  = tile * 16 + m15;
        int ptc = pt < M ? pt : (M - 1);
        const float* pp = points + ((size_t)b * M + ptc) * 3;
        float px = pp[0], py = pp[1], pz = pp[2];
        // B matrix (4x16): column = (x,y,z,1). lanes 0-15: K=0,1 ; lanes 16-31: K=2,3
        v2f bmat;
        bmat[0] = hi ? pz : px;
        bmat[1] = hi ? 1.0f : py;

        // prefetch the next tile this wave will process (global_prefetch_b8)
        int ntile = tile + waveStride;
        if (ntile < tilesPerBatch) {
            __builtin_prefetch(points + ((size_t)b * M + ntile * 16 + m15) * 3, 0, 0);
        }

        float acc = INFINITY;
        for (int i = 0; i < nPairs; ++i) {
            // A matrix (16x4): rows 0..2 = prim 2i affine, rows 8..10 = prim 2i+1 affine
            int aprim = 2 * i + (m15 >> 3);
            v2f amat;
            amat[0] = rowMask * prm[aprim][aoff];
            amat[1] = rowMask * prm[aprim][aoff + 1];
            v8f c = {};
            v8f d = __builtin_amdgcn_wmma_f32_16x16x4_f32(
                        /*neg_a=*/false, amat, /*neg_b=*/false, bmat,
                        /*c_mod=*/(short)0, c, /*reuse_a=*/false, /*reuse_b=*/false);

            // lane (0-15): d[0..2] = X,Y,Z of point m15 under prim 2i   (M=0..2)
            // lane (16-31): d[0..2] = same point under prim 2i+1        (M=8..10)
            const float* __restrict__ P = prm[2 * i + hi];
            float X = d[0], Y = d[1], Z = d[2];

            float x = sgnf(X) * fmaxf(fabsf(X), SQ_EPS);       // _sgn_clamp
            float y = sgnf(Y) * fmaxf(fabsf(Y), SQ_EPS);
            float z = sgnf(Z) * fmaxf(fabsf(Z), SQ_EPS);

            float pfx = P[15] * z;                              // safe_mul(kx/sz, z)
            pfx = sgnf(pfx) * clipEB(fabsf(pfx)) + 1.0f;
            float fx = sgnf(pfx) * fmaxf(fabsf(pfx), SQ_EPS);   // _sgn_clamp
            float pfy = P[16] * z;
            pfy = sgnf(pfy) * clipEB(fabsf(pfy)) + 1.0f;
            float fy = sgnf(pfy) * fmaxf(fabsf(pfy), SQ_EPS);
            x = x * frcp(fx);                                   // |fx| >= EPS: fast rcp safe
            y = y * frcp(fy);

            float u1 = x * P[12], u2 = y * P[13], u3 = z * P[14];
            // term = safe_pow(safe_pow(u,2), p) = sgn(u)*clip(clip(|u|)^2)^p
            float q1 = clipEB(fabsf(u1));
            float t1 = sgnf(u1) * powp(clipEB(q1 * q1), P[17]); // ^(1/e2)
            float q2 = clipEB(fabsf(u2));
            float t2 = sgnf(u2) * powp(clipEB(q2 * q2), P[17]); // ^(1/e2)
            float q3 = clipEB(fabsf(u3));
            float t3 = sgnf(u3) * powp(clipEB(q3 * q3), P[18]); // ^(1/e1)

            float s12 = t1 + t2;
            float g = sgnf(s12) * powp(clipEB(fabsf(s12)), P[19]); // ^(e2/e1)
            float h = g + t3;
            float f = sgnf(h) * powp(clipEB(fabsf(h)), P[20]);     // ^(-e1/2)
            float sdf = 1.0f - f;
            sdf = (P[21] > 0.5f) ? sdf : INFINITY;                 // exist mask
            acc = fminf(acc, sdf);
        }
        // fold the two primitives held by lane L and L+16
        float other = __shfl_xor(acc, 16, 32);
        acc = fminf(acc, other);
        if (hi == 0 && pt < M) out[(size_t)b * M + pt] = acc;
    }
}

extern "C" void kernel_launch(void* const* d_in, const int* in_sizes, int n_in,
                              void* d_out, int out_size, void* d_ws, size_t ws_size,
                              hipStream_t stream) {
    (void)n_in; (void)d_ws; (void)ws_size;
    const float* points   = (const float*)d_in[0];
    const float* rscale   = (const float*)d_in[1];
    const float* rexps    = (const float*)d_in[2];
    const float* rrot     = (const float*)d_in[3];
    const float* trans    = (const float*)d_in[4];
    const float* rtap     = (const float*)d_in[5];
    const int*   emask    = (const int*)d_in[6];
    float* out = (float*)d_out;

    const int B = 16;                       // fixed by the reference problem
    const int N = in_sizes[6] / B;          // 32
    const int M = out_size / B;             // 100000
    const int tilesPerBatch = (M + 15) / 16;

    dim3 block(256);                        // 8 wave32 per block
    dim3 grid(128, B);                      // grid-stride over 16-point tiles per batch
    hipLaunchKernelGGL(sq_sdf_kernel, grid, block, 0, stream,
                       points, rscale, rexps, rrot, trans, rtap, emask, out,
                       M, N, tilesPerBatch);
}